// GCNNet_1056561954979
// MI455X (gfx1250) — compile-verified
//
#include <hip/hip_runtime.h>
#include <math.h>

// ---------------------------------------------------------------------------
// GCN link-prediction pipeline for MI455X (gfx1250, wave32).
// Bottleneck = edge scatter (native f32 atomics); GEMMs use fp32 WMMA
// (V_WMMA_F32_16X16X4_F32) for exact fp32 numerics at negligible cost.
// ---------------------------------------------------------------------------

typedef __attribute__((ext_vector_type(2))) float v2f;
typedef __attribute__((ext_vector_type(8))) float v8f;

#define FDIM 128   // input / hidden width
#define DDIM 32    // relation embedding width

// ---------------- utility: zero a float buffer ------------------------------
__global__ void zero_kernel(float* __restrict__ p, long long n) {
    long long i = (long long)blockIdx.x * blockDim.x + threadIdx.x;
    if (i < n) p[i] = 0.0f;
}

// ---------------- degree counting (float counters, native atomics) ----------
__global__ void degree_kernel(const int* __restrict__ src, const int* __restrict__ dst,
                              float* __restrict__ deg_out, float* __restrict__ deg_in, int E) {
    int e = blockIdx.x * blockDim.x + threadIdx.x;
    if (e < E) {
        unsafeAtomicAdd(&deg_out[src[e]], 1.0f);
        unsafeAtomicAdd(&deg_in[dst[e]], 1.0f);
    }
}

// in-place deg -> rsqrt(max(deg,1))
__global__ void rsqrt_kernel(float* __restrict__ d, int n) {
    int i = blockIdx.x * blockDim.x + threadIdx.x;
    if (i < n) d[i] = rsqrtf(fmaxf(d[i], 1.0f));
}

// ---------------- edge scatter: agg[dst] += x[src] * rs_out[src] ------------
// 32 threads per edge, 4 consecutive floats per thread (float4 gather).
__global__ void scatter_kernel(const float* __restrict__ xin, const float* __restrict__ rs_out,
                               const int* __restrict__ src, const int* __restrict__ dst,
                               float* __restrict__ agg, long long total) {
    long long t = (long long)blockIdx.x * blockDim.x + threadIdx.x;
    if (t >= total) return;
    int e = (int)(t >> 5);
    int c = ((int)t & 31) * 4;
    int s = src[e];
    int d = dst[e];
    float sc = rs_out[s];
    float4 v = *(const float4*)(xin + (long long)s * FDIM + c);
    float* ap = agg + (long long)d * FDIM + c;
    unsafeAtomicAdd(ap + 0, v.x * sc);
    unsafeAtomicAdd(ap + 1, v.y * sc);
    unsafeAtomicAdd(ap + 2, v.z * sc);
    unsafeAtomicAdd(ap + 3, v.w * sc);
}

// ---------------- GCN layer GEMM: out = tanh((agg .* rs_in) @ W + b) --------
// W is [128 x 128] row-major (k-major). Block = 256 = 8 waves; each wave owns
// one 16x16 output tile (grid.x picks the 16-row tile, wave id picks columns).
// fp32 WMMA: A-frag lane L holds rows m0+(L&15), K pair {kg, kg+1}, kg=2*(L>>4).
__global__ __launch_bounds__(256) void gcn_gemm_kernel(
        const float* __restrict__ agg, const float* __restrict__ rs_in,
        const float* __restrict__ W, const float* __restrict__ bias,
        float* __restrict__ out) {
    const int lane = threadIdx.x & 31;
    const int wave = threadIdx.x >> 5;
    const int l15  = lane & 15;
    const int kg   = (lane >> 4) << 1;          // 0 or 2
    const int m0   = blockIdx.x * 16;
    const int n0   = wave * 16;
    const int arow = m0 + l15;
    const int bcol = n0 + l15;
    const float rs = rs_in[arow];
    const float* arp = agg + (long long)arow * FDIM;

    v8f c = {};
    #pragma unroll 4
    for (int k = 0; k < FDIM; k += 4) {
        v2f a, b;
        a.x = arp[k + kg]     * rs;
        a.y = arp[k + kg + 1] * rs;
        b.x = W[(long long)(k + kg)     * FDIM + bcol];
        b.y = W[(long long)(k + kg + 1) * FDIM + bcol];
        c = __builtin_amdgcn_wmma_f32_16x16x4_f32(
                false, a, false, b, (short)0, c, false, false);
    }
    const float bv = bias[bcol];
    const int rbase = m0 + ((lane >> 4) << 3);  // lanes 16-31 hold rows m0+8..15
    #pragma unroll
    for (int g = 0; g < 8; ++g)
        out[(long long)(rbase + g) * FDIM + bcol] = tanhf(c[g] + bv);
}

// ---------------- z = concat(x, input_feat) @ W_lin  (K=256, Nout=32) -------
// Block = 64 threads = 2 waves covering the 2 column tiles; no bias/activation.
__global__ __launch_bounds__(64) void lin_gemm_kernel(
        const float* __restrict__ x, const float* __restrict__ feat,
        const float* __restrict__ Wl, float* __restrict__ z) {
    const int lane = threadIdx.x & 31;
    const int wave = threadIdx.x >> 5;
    const int l15  = lane & 15;
    const int kg   = (lane >> 4) << 1;
    const int m0   = blockIdx.x * 16;
    const int bcol = wave * 16 + l15;
    const int arow = m0 + l15;

    v8f c = {};
    #pragma unroll 4
    for (int k = 0; k < 2 * FDIM; k += 4) {
        // k is a multiple of 4 -> the whole 4-wide K group sits in one half
        const float* A = (k < FDIM) ? (x + (long long)arow * FDIM)
                                    : (feat + (long long)arow * FDIM - FDIM);
        v2f a, b;
        a.x = A[k + kg];
        a.y = A[k + kg + 1];
        b.x = Wl[(long long)(k + kg)     * DDIM + bcol];
        b.y = Wl[(long long)(k + kg + 1) * DDIM + bcol];
        c = __builtin_amdgcn_wmma_f32_16x16x4_f32(
                false, a, false, b, (short)0, c, false, false);
    }
    const int rbase = m0 + ((lane >> 4) << 3);
    #pragma unroll
    for (int g = 0; g < 8; ++g)
        z[(long long)(rbase + g) * DDIM + bcol] = c[g];
}

// ---------------- row L2 normalize (in place), eps = 1e-12 ------------------
__global__ void l2norm_kernel(float* __restrict__ z, int n) {
    int row = blockIdx.x * blockDim.x + threadIdx.x;
    if (row >= n) return;
    float* p = z + (long long)row * DDIM;
    float s = 0.0f;
    #pragma unroll
    for (int d = 0; d < DDIM; ++d) s += p[d] * p[d];
    float inv = 1.0f / fmaxf(sqrtf(s), 1e-12f);
    #pragma unroll
    for (int d = 0; d < DDIM; ++d) p[d] *= inv;
}

// ---------------- TransE-style scoring --------------------------------------
__global__ void score_kernel(const float* __restrict__ zn, const float* __restrict__ rel,
                             const int* __restrict__ psrc, const int* __restrict__ pdst,
                             const int* __restrict__ ndst, const int* __restrict__ et,
                             float* __restrict__ out, int P) {
    int p = blockIdx.x * blockDim.x + threadIdx.x;
    if (p >= P) return;
    const float* r  = rel + (long long)et[p]   * DDIM;
    const float* h  = zn  + (long long)psrc[p] * DDIM;
    const float* t  = zn  + (long long)pdst[p] * DDIM;
    const float* tn = zn  + (long long)ndst[p] * DDIM;

    float rv[DDIM];
    float rn2 = 0.0f;
    #pragma unroll
    for (int d = 0; d < DDIM; ++d) { rv[d] = r[d]; rn2 += rv[d] * rv[d]; }
    float rscale = 1.0f / fmaxf(sqrtf(rn2), 1.0f);   // max_norm=1 renorm

    float sp = 0.0f, sn = 0.0f;
    #pragma unroll
    for (int d = 0; d < DDIM; ++d) {
        float hr = h[d] + rv[d] * rscale;
        float dp = hr - t[d];  sp += dp * dp;
        float dn = hr - tn[d]; sn += dn * dn;
    }
    out[p]     = sqrtf(sp);
    out[P + p] = sqrtf(sn);
}

// ---------------------------------------------------------------------------
extern "C" void kernel_launch(void* const* d_in, const int* in_sizes, int n_in,
                              void* d_out, int out_size, void* d_ws, size_t ws_size,
                              hipStream_t stream) {
    const float* input_feat = (const float*)d_in[0];
    const float* W1     = (const float*)d_in[1];
    const float* b1     = (const float*)d_in[2];
    const float* W2     = (const float*)d_in[3];
    const float* b2     = (const float*)d_in[4];
    const float* W_lin  = (const float*)d_in[5];
    const float* rel    = (const float*)d_in[6];
    const int*   src    = (const int*)d_in[7];
    const int*   dst    = (const int*)d_in[8];
    const int*   psrc   = (const int*)d_in[9];
    const int*   pdst   = (const int*)d_in[10];
    const int*   ndst   = (const int*)d_in[11];
    const int*   et     = (const int*)d_in[12];

    const int N = in_sizes[0] / FDIM;   // 50000 (divisible by 16)
    const int E = in_sizes[7];          // 800000
    const int P = in_sizes[9];          // 200000

    // workspace layout (floats)
    float* ws     = (float*)d_ws;
    float* rs_out = ws;                              // N  (deg_out -> rsqrt)
    float* rs_in  = rs_out + N;                      // N  (deg_in  -> rsqrt)
    float* agg    = rs_in + N;                       // N*128
    float* x      = agg + (long long)N * FDIM;       // N*128
    float* z      = x   + (long long)N * FDIM;       // N*32

    const int TB = 256;
    const long long nAgg = (long long)N * FDIM;
    const long long nScat = (long long)E * 32;
    const int rowTiles = N / 16;

    // degrees -> rsqrt(max(deg,1))
    zero_kernel<<<(int)((2LL * N + TB - 1) / TB), TB, 0, stream>>>(rs_out, 2LL * N);
    degree_kernel<<<(E + TB - 1) / TB, TB, 0, stream>>>(src, dst, rs_out, rs_in, E);
    rsqrt_kernel<<<(2 * N + TB - 1) / TB, TB, 0, stream>>>(rs_out, 2 * N);

    // ---- GCN layer 1 ----
    zero_kernel<<<(int)((nAgg + TB - 1) / TB), TB, 0, stream>>>(agg, nAgg);
    scatter_kernel<<<(int)((nScat + TB - 1) / TB), TB, 0, stream>>>(
        input_feat, rs_out, src, dst, agg, nScat);
    gcn_gemm_kernel<<<rowTiles, 256, 0, stream>>>(agg, rs_in, W1, b1, x);

    // ---- GCN layer 2 ----
    zero_kernel<<<(int)((nAgg + TB - 1) / TB), TB, 0, stream>>>(agg, nAgg);
    scatter_kernel<<<(int)((nScat + TB - 1) / TB), TB, 0, stream>>>(
        x, rs_out, src, dst, agg, nScat);
    gcn_gemm_kernel<<<rowTiles, 256, 0, stream>>>(agg, rs_in, W2, b2, x);

    // ---- linear + normalize + score ----
    lin_gemm_kernel<<<rowTiles, 64, 0, stream>>>(x, input_feat, W_lin, z);
    l2norm_kernel<<<(N + TB - 1) / TB, TB, 0, stream>>>(z, N);
    score_kernel<<<(P + TB - 1) / TB, TB, 0, stream>>>(
        z, rel, psrc, pdst, ndst, et, (float*)d_out, P);
}